// LoCon1d_59021440582064
// MI455X (gfx1250) — compile-verified
//
#include <hip/hip_runtime.h>
#include <hip/hip_bf16.h>

// LoCon1d: position-specific conv1d as 1024 independent 16x64x192 GEMMs,
// computed with V_WMMA_F32_16X16X4_F32 on gfx1250 (wave32).
//
// Shapes (all f32):
//   input  (B=16, Cin=64, S=1024)
//   weight (Cout=64, Cin=64, S=1024, K=3)
//   bias   (Cout=64, S=1024)
//   out    (B=16, Cout=64, S=1024)
//
// Memory-bound: ~58.6 MB moved once, 0.40 GFLOP -> 6.9 FLOP/B; the f32 WMMA
// path supplies the ~160 TFLOP/s needed to stay at the 23.3 TB/s roofline.

typedef __attribute__((ext_vector_type(2))) float v2f;
typedef __attribute__((ext_vector_type(8))) float v8f;

#define CIN   64
#define COUT  64
#define SLEN  1024
#define KW    3

__global__ __launch_bounds__(128)
void locon1d_wmma_f32(const float* __restrict__ in,    // (16,64,1024)
                      const float* __restrict__ wgt,   // (64,64,1024,3)
                      const float* __restrict__ bias,  // (64,1024)
                      float* __restrict__ out)         // (16,64,1024)
{
    const int s    = blockIdx.x;            // position, 0..1023
    const int lane = threadIdx.x & 31;
    const int wave = threadIdx.x >> 5;      // 0..3 -> Cout tile
    const int o0   = wave * 16;
    const int half = lane >> 4;             // selects K pair within the chunk
    const int lid  = lane & 15;             // A: M=batch; B/D: N=cout-in-tile
    const int cc0  = half * 2;              // channel offset of this lane's pair

    v8f acc = {};

    // B fragment: lane's column is cout o = o0 + lid.
    const float* wbase = wgt + (size_t)(o0 + lid) * (CIN * SLEN * KW) + (size_t)s * KW;
    // A fragment: lane's row is batch b = lid.
    const float* ibase = in + (size_t)lid * (CIN * SLEN);

    // Contraction ordered k-major (ck = k*64 + c): three k-sections of 16
    // WMMA steps each. Zero-padding is exact section skipping: an
    // out-of-range spos makes the whole section's A zero, so drop it with a
    // single wave-uniform branch (taken only for s==0 / s==1023). EXEC stays
    // all-ones around every WMMA.
    #pragma unroll
    for (int k = 0; k < KW; ++k) {
        const int spos = s + k - 1;                  // uniform per section
        if (spos < 0 || spos >= SLEN) continue;      // contributes exactly zero

        const float* iptr = ibase + spos;            // + c*SLEN per channel
        const float* wptr = wbase + k;               // + c*SLEN*KW per channel

        // Stage 1: issue all 64 loads for this section (compiler can clause
        // them and stage the waits), fragments land in register arrays.
        v2f af[16], bf[16];
        #pragma unroll
        for (int j = 0; j < 16; ++j) {
            const int cc = j * 4 + cc0;              // this lane's first channel
            // A 16x4 f32 layout (ISA 7.12.2): v0 = A[m][kk+2*half], v1 = +1.
            af[j].x = iptr[(size_t)cc * SLEN];
            af[j].y = iptr[(size_t)(cc + 1) * SLEN];
            // B 4x16 f32 layout (mirrored): v0 = B[kk+2*half][n], v1 = +1.
            bf[j].x = wptr[(size_t)cc * (SLEN * KW)];
            bf[j].y = wptr[(size_t)(cc + 1) * (SLEN * KW)];
        }

        // Stage 2: 16 chained WMMAs consume the staged fragments while the
        // remaining loads drain.
        #pragma unroll
        for (int j = 0; j < 16; ++j) {
            // (neg_a, A, neg_b, B, c_mod, C, reuse_a, reuse_b)
            acc = __builtin_amdgcn_wmma_f32_16x16x4_f32(
                false, af[j], false, bf[j], (short)0, acc, false, false);
        }
    }

    // bias depends on (o, s) only -> one scalar per lane, added to all 8 rows.
    const float bv = bias[(size_t)(o0 + lid) * SLEN + s];
    #pragma unroll
    for (int r = 0; r < 8; ++r) acc[r] += bv;

    // D 16x16 f32 layout: VGPR r, lanes 0-15 -> (M=r, N=lane);
    //                               lanes 16-31 -> (M=r+8, N=lane-16).
    const int mbase = half * 8;
    float* obase = out + (size_t)(o0 + lid) * SLEN + s;   // fixed (o, s)
    #pragma unroll
    for (int r = 0; r < 8; ++r) {
        obase[(size_t)(mbase + r) * (COUT * SLEN)] = acc[r];  // vary batch
    }
}

extern "C" void kernel_launch(void* const* d_in, const int* in_sizes, int n_in,
                              void* d_out, int out_size, void* d_ws, size_t ws_size,
                              hipStream_t stream) {
    const float* in   = (const float*)d_in[0];   // (16,64,1024)
    const float* wgt  = (const float*)d_in[1];   // (64,64,1024,3)
    const float* bias = (const float*)d_in[2];   // (64,1024)
    float*       out  = (float*)d_out;           // (16,64,1024)

    // One block per position s; 4 waves cover the 64 Cout columns.
    locon1d_wmma_f32<<<SLEN, 128, 0, stream>>>(in, wgt, bias, out);
}